// AAGNN_66322884985284
// MI455X (gfx1250) — compile-verified
//
#include <hip/hip_runtime.h>
#include <stdint.h>

typedef float v2f __attribute__((ext_vector_type(2)));
typedef float v4f __attribute__((ext_vector_type(4)));
typedef float v8f __attribute__((ext_vector_type(8)));
typedef int   v4i __attribute__((vector_size(16)));   // matches async builtin param type

#define N_NODES 16384
#define FEAT    128
#define HID     64
#define KB      32               // K-tile per LDS stage
#define NPAIR   (KB / 2)         // 16 pair-rows per tile
#define PSTR    80               // pair-row stride in float2; 80*2 dwords = 32 mod 64 banks
#define NTILES  (N_NODES / KB)   // 512

// ---------------------------------------------------------------------------
// CDNA5 async global->LDS copy (ASYNCcnt path), with sync fallback.
// ---------------------------------------------------------------------------
#if defined(__AMDGCN__) && __has_builtin(__builtin_amdgcn_global_load_async_to_lds_b128)
#define HAVE_ASYNC 1
#else
#define HAVE_ASYNC 0
#endif

#if __has_builtin(__builtin_amdgcn_s_wait_asynccnt)
#define WAIT_ASYNC(n) __builtin_amdgcn_s_wait_asynccnt(n)
#elif defined(__AMDGCN__)
#define WAIT_ASYNC(n) asm volatile("s_wait_asynccnt %0" ::"n"(n) : "memory")
#else
#define WAIT_ASYNC(n)
#endif

#if HAVE_ASYNC
typedef __attribute__((address_space(1))) v4i* as1_v4i_p;
typedef __attribute__((address_space(3))) v4i* as3_v4i_p;
// 16B/lane per call; INST_OFFSET applies to BOTH global and LDS addresses
// (ISA async pseudocode), so 4 calls cover 64 contiguous bytes on each side.
static __device__ __forceinline__ void async_stage64B(const void* g, void* l) {
  as1_v4i_p gp = (as1_v4i_p)(uintptr_t)g;
  as3_v4i_p lp = (as3_v4i_p)(uint32_t)(uintptr_t)l;
  __builtin_amdgcn_global_load_async_to_lds_b128(gp, lp, 0, 0);
  __builtin_amdgcn_global_load_async_to_lds_b128(gp, lp, 16, 0);
  __builtin_amdgcn_global_load_async_to_lds_b128(gp, lp, 32, 0);
  __builtin_amdgcn_global_load_async_to_lds_b128(gp, lp, 48, 0);
}
#endif

// 4 column-tile WMMAs for one K-substep; B comes pre-paired from LDS as an
// aligned ds_load_b64 (v2f) -- no operand shuffles.
#define WMMA_SUBSTEP(a_, brow_, n_)                                            \
  {                                                                            \
    acc0 = __builtin_amdgcn_wmma_f32_16x16x4_f32(false, (a_), false,           \
             (brow_)[(n_)],      (short)0, acc0, false, false);                \
    acc1 = __builtin_amdgcn_wmma_f32_16x16x4_f32(false, (a_), false,           \
             (brow_)[(n_) + 16], (short)0, acc1, false, false);                \
    acc2 = __builtin_amdgcn_wmma_f32_16x16x4_f32(false, (a_), false,           \
             (brow_)[(n_) + 32], (short)0, acc2, false, false);                \
    acc3 = __builtin_amdgcn_wmma_f32_16x16x4_f32(false, (a_), false,           \
             (brow_)[(n_) + 48], (short)0, acc3, false, false);                \
  }

// ===========================================================================
// Kernel 2 (hot, emitted first for disasm visibility):
//   out = relu(dn * (adj @ S) + b), S consumed in packed-pair form.
// 4 waves/block, 16-row strip/wave, KB=32 K-tiles double-buffered in LDS via
// async DMA. adj streams nontemporally (1 GiB single-use > L2).
// ===========================================================================
__global__ __launch_bounds__(128)
void gcn_agg_kernel(const float* __restrict__ adj, const float* __restrict__ Sp,
                    const float* __restrict__ dn, const float* __restrict__ bias,
                    float* __restrict__ out) {
  __shared__ v2f sB[2][NPAIR * PSTR];          // 2 x 10 KB
  const int lane = threadIdx.x & 31;
  const int wave = threadIdx.x >> 5;
  const int hi   = lane >> 4;
  const int n    = lane & 15;
  const int rowBase = blockIdx.x * 64 + wave * 16;
  const float* arow = adj + (size_t)(rowBase + n) * N_NODES;

  const float bn0 = bias[n];
  const float bn1 = bias[n + 16];
  const float bn2 = bias[n + 32];
  const float bn3 = bias[n + 48];

  // staging: thread -> pair-row tp (0..15), 8 float2 columns from tc
  const int tp = threadIdx.x >> 3;
  const int tc = (threadIdx.x & 7) * 8;
  const v2f* gstage = (const v2f*)Sp + (size_t)tp * HID + tc;  // + tile*NPAIR*HID
  v2f* lstage0 = &sB[0][tp * PSTR + tc];
  v2f* lstage1 = &sB[1][tp * PSTR + tc];

  v8f acc0 = {}, acc1 = {}, acc2 = {}, acc3 = {};

#if HAVE_ASYNC
  async_stage64B(gstage, lstage0);                     // prefetch tile 0
#endif

  for (int t = 0; t < NTILES; ++t) {
    const v2f* buf = (t & 1) ? sB[1] : sB[0];

#if HAVE_ASYNC
    __syncthreads();  // everyone done with the buffer we are about to refill
    if (t + 1 < NTILES) {
      async_stage64B(gstage + (size_t)(t + 1) * (NPAIR * HID),
                     (t & 1) ? lstage0 : lstage1);
      WAIT_ASYNC(4);  // oldest 4 (tile t) complete; tile t+1 stays in flight
    } else {
      WAIT_ASYNC(0);
    }
    __syncthreads();  // tile t visible to all waves
#else
    __syncthreads();
    {
      v2f* lb = (t & 1) ? lstage1 : lstage0;
      const v2f* g = gstage + (size_t)t * (NPAIR * HID);
#pragma unroll
      for (int j = 0; j < 4; ++j)
        *(v4f*)((float*)lb + 4 * j) = *(const v4f*)((const float*)g + 4 * j);
    }
    __syncthreads();
#endif

    const int k0 = t * KB;
#pragma unroll
    for (int s = 0; s < 8; ++s) {
      v2f a = __builtin_nontemporal_load((const v2f*)(arow + k0 + 4 * s + 2 * hi));
      const v2f* brow = buf + (2 * s + hi) * PSTR;
      WMMA_SUBSTEP(a, brow, n);
    }
  }

#pragma unroll
  for (int i = 0; i < 8; ++i) {
    const int row = rowBase + i + 8 * hi;
    const float d = dn[row];
    float* o = out + (size_t)row * HID + n;
    float v0 = d * acc0[i] + bn0;
    float v1 = d * acc1[i] + bn1;
    float v2 = d * acc2[i] + bn2;
    float v3 = d * acc3[i] + bn3;
    o[0]  = v0 > 0.0f ? v0 : 0.0f;
    o[16] = v1 > 0.0f ? v1 : 0.0f;
    o[32] = v2 > 0.0f ? v2 : 0.0f;
    o[48] = v3 > 0.0f ? v3 : 0.0f;
  }
}

// ===========================================================================
// Kernel 1:  Sp = pack(x @ W),  Sp[p][c] = { (xW)[2p][c], (xW)[2p+1][c] }.
// Same packed-pair LDS scheme (W rows interleaved at stage time), so the
// WMMA-B feed is also a single aligned ds_load_b64 here.
// ===========================================================================
__global__ __launch_bounds__(128)
void gcn_xw_kernel(const float* __restrict__ x, const float* __restrict__ W,
                   float* __restrict__ Sp) {
  __shared__ v2f sW[NPAIR * PSTR];             // 10 KB
  const int lane = threadIdx.x & 31;
  const int wave = threadIdx.x >> 5;
  const int hi   = lane >> 4;
  const int n    = lane & 15;
  const int rowBase = blockIdx.x * 64 + wave * 16;
  const float* arow = x + (size_t)(rowBase + n) * FEAT;

  const int tp = threadIdx.x >> 3;
  const int tc = (threadIdx.x & 7) * 8;
  v2f* lstage = &sW[tp * PSTR + tc];

  v8f acc0 = {}, acc1 = {}, acc2 = {}, acc3 = {};

  for (int k0 = 0; k0 < FEAT; k0 += KB) {
    __syncthreads();
    {
      const float* r0 = W + (size_t)(k0 + 2 * tp)     * HID + tc;
      const float* r1 = W + (size_t)(k0 + 2 * tp + 1) * HID + tc;
      v2f tmp[8];
#pragma unroll
      for (int j = 0; j < 8; ++j) { tmp[j].x = r0[j]; tmp[j].y = r1[j]; }
#pragma unroll
      for (int j = 0; j < 4; ++j)
        *(v4f*)((float*)lstage + 4 * j) = *(const v4f*)((const float*)&tmp[2 * j]);
    }
    __syncthreads();
#pragma unroll
    for (int s = 0; s < 8; ++s) {
      v2f a = *(const v2f*)(arow + k0 + 4 * s + 2 * hi);
      const v2f* brow = sW + (2 * s + hi) * PSTR;
      WMMA_SUBSTEP(a, brow, n);
    }
  }

  // store pair-packed: float index = (row>>1)*128 + col*2 + (row&1)
#pragma unroll
  for (int i = 0; i < 8; ++i) {
    const int row = rowBase + i + 8 * hi;
    float* o = Sp + (size_t)(row >> 1) * 128 + (row & 1) + 2 * n;
    o[0]  = acc0[i];   // col n
    o[32] = acc1[i];   // col n+16
    o[64] = acc2[i];   // col n+32
    o[96] = acc3[i];   // col n+48
  }
}

// ===========================================================================
extern "C" void kernel_launch(void* const* d_in, const int* in_sizes, int n_in,
                              void* d_out, int out_size, void* d_ws, size_t ws_size,
                              hipStream_t stream) {
  const float* x   = (const float*)d_in[0];   // [16384,128]
  const float* adj = (const float*)d_in[1];   // [16384,16384]
  const float* dn  = (const float*)d_in[2];   // [16384,1]
  const float* W   = (const float*)d_in[3];   // [128,64]
  const float* b   = (const float*)d_in[4];   // [64]
  float* out = (float*)d_out;                 // [16384,64]
  float* Sp  = (float*)d_ws;                  // 4 MB packed support matrix

  dim3 block(128);
  dim3 grid(N_NODES / 64);

  gcn_xw_kernel<<<grid, block, 0, stream>>>(x, W, Sp);
  gcn_agg_kernel<<<grid, block, 0, stream>>>(adj, Sp, dn, b, out);
}